// L1Loss_40286793236680
// MI455X (gfx1250) — compile-verified
//
#include <hip/hip_runtime.h>

typedef float v2f __attribute__((ext_vector_type(2)));
typedef float v8f __attribute__((ext_vector_type(8)));

#define PLANE 16384       // 128*128 elements per (b,c) plane
#define THREADS 256
#define ITERS (PLANE / (THREADS * 4))   // 16 float4 loads per thread per tensor

// One block per (b,c) plane. Streams pre/gt/mask with b128 loads, accumulates
// per-lane (masked L1 sum, nonzero count), reduces the wave with a single
// v_wmma_f32_16x16x4_f32, combines 8 waves via LDS, writes per-plane
// sum / max(cnt,1) to partials[plane].
__global__ __launch_bounds__(THREADS) void masked_l1_plane_kernel(
    const float* __restrict__ pre, const float* __restrict__ gt,
    const float* __restrict__ mask, float* __restrict__ partials)
{
    const int tid   = threadIdx.x;
    const int plane = blockIdx.x;
    const size_t base = (size_t)plane * PLANE;

    const float4* __restrict__ p4 = reinterpret_cast<const float4*>(pre  + base);
    const float4* __restrict__ g4 = reinterpret_cast<const float4*>(gt   + base);
    const float4* __restrict__ m4 = reinterpret_cast<const float4*>(mask + base);

    float s = 0.0f;   // Σ |pre-gt| * mask
    float c = 0.0f;   // Σ (mask != 0)

    #pragma unroll 4
    for (int it = 0; it < ITERS; ++it) {
        const int i = it * THREADS + tid;       // coalesced float4 index
        const float4 p = p4[i];
        const float4 g = g4[i];
        const float4 m = m4[i];
        s += fabsf(p.x - g.x) * m.x;
        s += fabsf(p.y - g.y) * m.y;
        s += fabsf(p.z - g.z) * m.z;
        s += fabsf(p.w - g.w) * m.w;
        c += (m.x != 0.0f) ? 1.0f : 0.0f;
        c += (m.y != 0.0f) ? 1.0f : 0.0f;
        c += (m.z != 0.0f) ? 1.0f : 0.0f;
        c += (m.w != 0.0f) ? 1.0f : 0.0f;
    }

    // ---- wave32 reduction via WMMA f32 16x16x4 ----
    // A layout (16x4 f32): lane l in 0..15 -> A[l][0]=a.x, A[l][1]=a.y
    //                      lane l in 16..31 -> A[l-16][2]=a.x, A[l-16][3]=a.y
    // B layout (4x16 f32): lane n holds column n%16; VGPR0 = K0 (lanes 0-15) /
    //                      K2 (lanes 16-31); VGPR1 = K1 / K3.
    // Column 0 selector (1,0,1,0) -> D[m][0] = s_m + s_{m+16}
    // Column 1 selector (0,1,0,1) -> D[m][1] = c_m + c_{m+16}
    const int lane = tid & 31;
    const int lh   = lane & 15;

    v2f a; a.x = s; a.y = c;
    v2f b; b.x = (lh == 0) ? 1.0f : 0.0f;
           b.y = (lh == 1) ? 1.0f : 0.0f;
    v8f acc = {};
    acc = __builtin_amdgcn_wmma_f32_16x16x4_f32(
        /*neg_a=*/false, a, /*neg_b=*/false, b,
        /*c_mod=*/(short)0, acc, /*reuse_a=*/false, /*reuse_b=*/false);

    // D (16x16 f32, 8 VGPRs): lane n holds column n, rows 0-7 (lanes 0-15)
    // or rows 8-15 (lanes 16-31). Sum own 8 rows, then add lane^16 partner:
    float t = acc[0] + acc[1] + acc[2] + acc[3] + acc[4] + acc[5] + acc[6] + acc[7];
    t += __shfl_xor(t, 16, 32);
    // lane 0 now holds the full-wave masked-L1 sum, lane 1 the full-wave count.

    __shared__ float lsum[THREADS / 32];
    __shared__ float lcnt[THREADS / 32];
    const int wave = tid >> 5;
    if (lane == 0) lsum[wave] = t;
    if (lane == 1) lcnt[wave] = t;
    __syncthreads();

    if (tid == 0) {
        float S = 0.0f, C = 0.0f;
        #pragma unroll
        for (int w = 0; w < THREADS / 32; ++w) { S += lsum[w]; C += lcnt[w]; }
        partials[plane] = S / fmaxf(C, 1.0f);
    }
}

// Deterministic final reduction of the 1024 per-plane losses; divides by
// batch_size (read from device). Fixed tree order -> bit-stable across replays.
__global__ __launch_bounds__(256) void final_reduce_kernel(
    const float* __restrict__ partials, const int* __restrict__ bs,
    float* __restrict__ out)
{
    __shared__ float red[256];
    const int tid = threadIdx.x;
    red[tid] = partials[tid] + partials[tid + 256] +
               partials[tid + 512] + partials[tid + 768];
    __syncthreads();
    #pragma unroll
    for (int s = 128; s > 0; s >>= 1) {
        if (tid < s) red[tid] += red[tid + s];
        __syncthreads();
    }
    if (tid == 0) out[0] = red[0] / (float)(*bs);
}

extern "C" void kernel_launch(void* const* d_in, const int* in_sizes, int n_in,
                              void* d_out, int out_size, void* d_ws, size_t ws_size,
                              hipStream_t stream) {
    const float* pre  = (const float*)d_in[0];
    const float* gt   = (const float*)d_in[1];
    const float* mask = (const float*)d_in[2];
    const int*   bs   = (const int*)d_in[3];

    float* partials = (float*)d_ws;   // 1024 floats of scratch

    const int n_planes = in_sizes[0] / PLANE;   // 16*64 = 1024
    masked_l1_plane_kernel<<<n_planes, THREADS, 0, stream>>>(pre, gt, mask, partials);
    final_reduce_kernel<<<1, 256, 0, stream>>>(partials, bs, (float*)d_out);
}